// Pseudo_random_Interleaver_3667902070959
// MI455X (gfx1250) — compile-verified
//
#include <hip/hip_runtime.h>

// Pseudo-random interleaver: out[b, l, :] = x[b, perm[b, l], :]
// B=64, L=4096, D=256, fp32.
//
// Pure streaming gather (perm is a permutation -> each row touched exactly
// once, zero reuse, zero FLOPs). Memory-bound: ~513 MiB total traffic,
// ~22 us floor at 23.3 TB/s. Optimal MI455X shape:
//   - b128 (float4) loads/stores, fully coalesced at 1 KiB row granularity
//   - non-temporal hints on both streams (no reuse; don't churn 192 MB L2)
//   - 32-bit index math only (row & ~0xFFF trick since L == 2^12)

typedef float v4f __attribute__((ext_vector_type(4)));

#define IL_B 64
#define IL_L 4096
#define IL_D 256
#define IL_D4 (IL_D / 4)              // 64 float4 per row
#define ROWS_PER_BLOCK 4              // 256 threads / 64 threads-per-row
#define TOTAL_ROWS (IL_B * IL_L)      // 262144

__global__ __launch_bounds__(256)
void interleave_gather_kernel(const v4f* __restrict__ x,
                              const int* __restrict__ perm,
                              v4f* __restrict__ out) {
    // Destination row this thread helps copy.
    const unsigned dst_row = blockIdx.x * ROWS_PER_BLOCK + (threadIdx.x >> 6);
    const unsigned col     = threadIdx.x & 63;           // float4 column in row

    // dst_row = b*4096 + l. perm value is the source l within the same batch,
    // so src_row = b*4096 + perm[dst_row] = (dst_row & ~0xFFF) | perm[dst_row].
    // (wave-uniform load: threadIdx.x>>6 is constant across a wave32)
    const unsigned p       = (unsigned)perm[dst_row];    // 0 <= p < 4096
    const unsigned src_row = (dst_row & 0xFFFFF000u) | p;

    // One b128 NT load + one b128 NT store per thread. Lanes of a wave cover
    // 32 consecutive float4s -> 512 B contiguous per VMEM instruction.
    const v4f v = __builtin_nontemporal_load(&x[(size_t)src_row * IL_D4 + col]);
    __builtin_nontemporal_store(v, &out[(size_t)dst_row * IL_D4 + col]);
}

extern "C" void kernel_launch(void* const* d_in, const int* in_sizes, int n_in,
                              void* d_out, int out_size, void* d_ws, size_t ws_size,
                              hipStream_t stream) {
    const v4f* x    = (const v4f*)d_in[0];   // float32 [B, L, D]
    const int* perm = (const int*)d_in[1];   // int32   [B, L]
    v4f*       out  = (v4f*)d_out;           // float32 [B, L, D]

    const int blocks = TOTAL_ROWS / ROWS_PER_BLOCK;  // 65536
    interleave_gather_kernel<<<blocks, 256, 0, stream>>>(x, perm, out);
}